// ProteinConcat_32908039422451
// MI455X (gfx1250) — compile-verified
//
#include <hip/hip_runtime.h>
#include <math.h>

typedef __attribute__((ext_vector_type(2))) float v2f;
typedef __attribute__((ext_vector_type(8))) float v8f;

#define B_     8
#define L_     2048
#define C_     4
#define D_     64
#define WIN_   9
#define NW_    (L_ - WIN_ + 1)      // 2040
#define OP_    256
#define KDIM   (D_ * 2 * WIN_)      // 1152
#define NCHUNK 32
#define LCHUNK (L_ / NCHUNK)        // 64
#define PART_SZ (B_ * NCHUNK * D_ * 18)   // floats
#define MPAD   16                   // padded M rows for WMMA

// ---------------------------------------------------------------------------
// Hardware tanh: CDNA5 has V_TANH_F32 as a single TRANS32 op.
// ---------------------------------------------------------------------------
__device__ __forceinline__ float fast_tanh(float x) {
#if __has_builtin(__builtin_amdgcn_tanh_f32)
    return __builtin_amdgcn_tanh_f32(x);
#elif __has_builtin(__builtin_amdgcn_tanhf)
    return __builtin_amdgcn_tanhf(x);
#else
    float r;
    // v_nop satisfies the TRANS-result hazard (1 op gap before use).
    asm("v_tanh_f32 %0, %1\n\tv_nop" : "=v"(r) : "v"(x));
    return r;
#endif
}

// ---------------------------------------------------------------------------
// Pass 1: windowed tanh-reduction. One block = (b, l-chunk). 256 threads =
// 64 d-lanes x 4 l-stripes. Interior chunks (1..30) use the mask-free path.
// ---------------------------------------------------------------------------
__global__ __launch_bounds__(256)
void pc_pass1_windowed(const float* __restrict__ p1, const float* __restrict__ p2,
                       const float* __restrict__ W, float* __restrict__ part) {
    __shared__ float wlds[C_ * D_ * WIN_];          // 2304 floats
    __shared__ float red[4][D_][18];                // 18 KB

    const int tid    = threadIdx.x;
    const int b      = blockIdx.x / NCHUNK;
    const int chunk  = blockIdx.x % NCHUNK;
    const int d      = tid & (D_ - 1);
    const int stripe = tid >> 6;

    for (int i = tid; i < C_ * D_ * WIN_; i += 256) wlds[i] = W[i];
    __syncthreads();

    float wreg[C_][WIN_];
#pragma unroll
    for (int c = 0; c < C_; ++c)
#pragma unroll
        for (int k = 0; k < WIN_; ++k)
            wreg[c][k] = wlds[(c * D_ + d) * WIN_ + k];

    float acc1[WIN_], acc2[WIN_];
#pragma unroll
    for (int k = 0; k < WIN_; ++k) { acc1[k] = 0.f; acc2[k] = 0.f; }

    const int l0 = chunk * LCHUNK;
    if (chunk != 0 && chunk != NCHUNK - 1) {
        // Interior: every l has all 9 windows valid (8 <= l <= 2039).
        for (int l = l0 + stripe; l < l0 + LCHUNK; l += 4) {
            const float* base1 = p1 + ((size_t)(b * L_ + l) * C_) * D_ + d;
            const float* base2 = p2 + ((size_t)(b * L_ + l) * C_) * D_ + d;
#pragma unroll
            for (int c = 0; c < C_; ++c) {
                const float x1 = base1[c * D_];
                const float x2 = base2[c * D_];
#pragma unroll
                for (int k = 0; k < WIN_; ++k) {
                    acc1[k] += fast_tanh(x1 * wreg[c][k]);
                    acc2[k] += fast_tanh(x2 * wreg[c][k]);
                }
            }
        }
    } else {
        // Edge chunks: mask out-of-range windows per (l, k).
        for (int l = l0 + stripe; l < l0 + LCHUNK; l += 4) {
            int kmin = l - (NW_ - 1); if (kmin < 0) kmin = 0;
            int kmax = (l < WIN_ - 1) ? l : (WIN_ - 1);
            const float* base1 = p1 + ((size_t)(b * L_ + l) * C_) * D_ + d;
            const float* base2 = p2 + ((size_t)(b * L_ + l) * C_) * D_ + d;
#pragma unroll
            for (int c = 0; c < C_; ++c) {
                const float x1 = base1[c * D_];
                const float x2 = base2[c * D_];
#pragma unroll
                for (int k = 0; k < WIN_; ++k) {
                    const float m = (k >= kmin && k <= kmax) ? 1.f : 0.f;
                    acc1[k] += m * fast_tanh(x1 * wreg[c][k]);
                    acc2[k] += m * fast_tanh(x2 * wreg[c][k]);
                }
            }
        }
    }

#pragma unroll
    for (int k = 0; k < WIN_; ++k) {
        red[stripe][d][k]     = acc1[k];
        red[stripe][d][9 + k] = acc2[k];
    }
    __syncthreads();

    if (stripe == 0) {
        float* dst = part + ((size_t)(b * NCHUNK + chunk) * D_ + d) * 18;
#pragma unroll
        for (int t = 0; t < 18; ++t)
            dst[t] = red[0][d][t] + red[1][d][t] + red[2][d][t] + red[3][d][t];
    }
}

// ---------------------------------------------------------------------------
// Pass 2: reduce 32 chunk partials, build act[b][d*18+t] = tanh(s1+s2 | s1*s2).
// Grid = 16 so rows 8..15 of the padded A matrix are physically zeroed,
// letting pass 3 skip the per-iteration row mask.
// ---------------------------------------------------------------------------
__global__ __launch_bounds__(576)
void pc_pass2_combine(const float* __restrict__ part, float* __restrict__ act) {
    const int b   = blockIdx.x;         // 0..15
    const int tid = threadIdx.x;        // 0..575
    const int d   = tid / WIN_;
    const int k   = tid % WIN_;

    float* dst = act + (size_t)b * KDIM + d * 18;
    if (b >= B_) {                      // zero-pad rows 8..15
        dst[k]     = 0.f;
        dst[9 + k] = 0.f;
        return;
    }

    float s1 = 0.f, s2 = 0.f;
    for (int ch = 0; ch < NCHUNK; ++ch) {
        const float* p = part + ((size_t)(b * NCHUNK + ch) * D_ + d) * 18;
        s1 += p[k];
        s2 += p[9 + k];
    }
    dst[k]     = fast_tanh(s1 + s2);
    dst[9 + k] = fast_tanh(s1 * s2);
}

// ---------------------------------------------------------------------------
// Pass 3: out(8x256) = act(16x1152) @ lin_w.T + lin_b via V_WMMA_F32_16X16X4_F32.
// One wave per 16-wide N tile; 288 k-steps of 4; pure load + wmma loop.
// ---------------------------------------------------------------------------
__global__ __launch_bounds__(32)
void pc_pass3_linear(const float* __restrict__ act, const float* __restrict__ lin_w,
                     const float* __restrict__ lin_b, float* __restrict__ out) {
    const int tile  = blockIdx.x;           // N tile: cols [tile*16, tile*16+16)
    const int lane  = threadIdx.x;          // 0..31
    const int mn    = lane & 15;            // row for A, col for B
    const int kbase = (lane >> 4) << 1;     // 0 or 2

    const float* arow = act   + (size_t)mn * KDIM;
    const float* brow = lin_w + (size_t)(tile * 16 + mn) * KDIM;

    v8f acc = {0.f, 0.f, 0.f, 0.f, 0.f, 0.f, 0.f, 0.f};
    for (int k0 = 0; k0 < KDIM; k0 += 4) {
        v2f a  = *(const v2f*)(arow + k0 + kbase);
        v2f bb = *(const v2f*)(brow + k0 + kbase);
        // 8 args: (neg_a, A, neg_b, B, c_mod, C, reuse_a, reuse_b)
        acc = __builtin_amdgcn_wmma_f32_16x16x4_f32(
            false, a, false, bb, (short)0, acc, false, false);
    }

    // D layout: VGPR j, lanes 0-15 -> row M=j, col N=lane. Store rows 0..7 only.
    if (lane < 16) {
        const int col = tile * 16 + lane;
        const float bias = lin_b[col];
#pragma unroll
        for (int j = 0; j < 8; ++j)
            out[(size_t)j * OP_ + col] = acc[j] + bias;
    }
}

// ---------------------------------------------------------------------------
extern "C" void kernel_launch(void* const* d_in, const int* in_sizes, int n_in,
                              void* d_out, int out_size, void* d_ws, size_t ws_size,
                              hipStream_t stream) {
    const float* p1    = (const float*)d_in[0];
    const float* p2    = (const float*)d_in[1];
    const float* W     = (const float*)d_in[2];
    const float* lin_w = (const float*)d_in[3];
    const float* lin_b = (const float*)d_in[4];

    float* part = (float*)d_ws;          // PART_SZ floats
    float* act  = part + PART_SZ;        // MPAD*KDIM floats (rows 8..15 zeroed)

    pc_pass1_windowed<<<B_ * NCHUNK, 256, 0, stream>>>(p1, p2, W, part);
    pc_pass2_combine<<<MPAD, 576, 0, stream>>>(part, act);
    pc_pass3_linear<<<OP_ / 16, 32, 0, stream>>>(act, lin_w, lin_b, (float*)d_out);
}